// NAGVAC_72078141161662
// MI455X (gfx1250) — compile-verified
//
#include <hip/hip_runtime.h>

typedef __attribute__((ext_vector_type(2))) float v2f;
typedef __attribute__((ext_vector_type(8))) float v8f;
typedef __attribute__((ext_vector_type(4))) int   v4i;

#define S_SAMPLES 1024
#define DLAT      4096
#define KC        128          // k-tile staged in LDS per step
#define LDS_STRIDE 132         // 128 + 4 pad -> conflict-free WMMA-layout reads
#define KCHUNKS   8            // k split across blocks for occupancy
#define KPC       (DLAT / KCHUNKS)      // 512 k per block
#define NTILES    (KPC / KC)            // 4 tiles per block

// ---- gfx1250 async global->LDS path (ASYNCcnt), builtin or inline-asm ------
#if defined(__has_builtin)
#  if __has_builtin(__builtin_amdgcn_global_load_async_to_lds_b128)
#    define USE_ASYNC_BUILTIN 1
#  endif
#  if __has_builtin(__builtin_amdgcn_s_wait_asynccnt)
#    define USE_WAIT_BUILTIN 1
#  endif
#endif

#ifdef USE_WAIT_BUILTIN
#  define ASYNC_WAIT(n) __builtin_amdgcn_s_wait_asynccnt(n)
#else
#  define ASYNC_WAIT(n) asm volatile("s_wait_asynccnt %0" :: "n"(n) : "memory")
#endif

typedef __attribute__((address_space(1))) v4i as1_v4i;   // global int4
typedef __attribute__((address_space(3))) v4i as3_v4i;   // LDS int4 (32-bit ptr)

__device__ __forceinline__ void async_copy_b128(const float* g, float* l) {
#ifdef USE_ASYNC_BUILTIN
    // Signature (from clang diagnostic): (global v4i* src, local v4i* dst, offset, cpol)
    __builtin_amdgcn_global_load_async_to_lds_b128(
        (as1_v4i*)(uintptr_t)g,
        (as3_v4i*)(unsigned)(uintptr_t)l,   // flat addr[31:0] == LDS offset
        0, 0);
#else
    // GV mode: vdst = LDS byte address, v[addr] = 64-bit global address
    asm volatile("global_load_async_to_lds_b128 %0, %1, off"
                 :: "v"((unsigned)(uintptr_t)l),
                    "v"((unsigned long long)(uintptr_t)g)
                 : "memory");
#endif
}

// Issue one 16x128-float tile (512 x 16B chunks, 2 per thread), coalesced.
__device__ __forceinline__ void stage_tile(const float* __restrict__ theta,
                                           float* ldsbuf, int tid,
                                           int row0, int k0) {
#pragma unroll
    for (int i = 0; i < 2; ++i) {
        int idx = tid + i * 256;       // 0..511
        int r   = idx >> 5;            // 32 x 16B chunks per row
        int kc  = idx & 31;
        const float* g = theta + (size_t)(row0 + r) * DLAT + (k0 + kc * 4);
        float*       l = ldsbuf + r * LDS_STRIDE + kc * 4;   // 16B aligned (528B row pitch)
        async_copy_b128(g, l);
    }
}

// WMMA row-sum over one staged tile: accA += sum d^2*w1, accB += sum d*w2.
__device__ __forceinline__ void compute_tile(const float* __restrict__ tilebuf,
                                             const float* __restrict__ mu,
                                             const float* __restrict__ w1,
                                             const float* __restrict__ w2,
                                             int m, int kh, int wave, int kgbase,
                                             v2f bones, v8f& accA, v8f& accB) {
    const int kbase = wave * 16;       // each wave owns a 16-wide k-stripe
#pragma unroll
    for (int kq = 0; kq < 16; kq += 4) {
        const int kk = kbase + kq + 2 * kh;   // this lane's K (A-layout)
        const int kg = kgbase + kk;
        float x0 = tilebuf[m * LDS_STRIDE + kk];
        float x1 = tilebuf[m * LDS_STRIDE + kk + 1];
        float d0 = x0 - mu[kg];
        float d1 = x1 - mu[kg + 1];
        float a0 = w1[kg], a1 = w1[kg + 1];
        float b0 = w2[kg], b1 = w2[kg + 1];
        v2f aA; aA[0] = d0 * d0 * a0; aA[1] = d1 * d1 * a1;
        v2f aB; aB[0] = d0 * b0;      aB[1] = d1 * b1;
        accA = __builtin_amdgcn_wmma_f32_16x16x4_f32(
            false, aA, false, bones, (short)0, accA, false, false);
        accB = __builtin_amdgcn_wmma_f32_16x16x4_f32(
            false, aB, false, bones, (short)0, accB, false, false);
    }
}

// ---------------------------------------------------------------------------
// Kernel 1: O(D) prep. c = softplus(log_c)+1e-4; w1 = 1/c^2; w2 = b/c^2;
// params = { 1/(1+t), D*log(2pi) + sum(log c^2) + log1p(t) }, t = sum(b^2/c^2)
// ---------------------------------------------------------------------------
__global__ __launch_bounds__(256) void nagvac_prep(const float* __restrict__ b,
                                                   const float* __restrict__ log_c,
                                                   float* __restrict__ w1,
                                                   float* __restrict__ w2,
                                                   float* __restrict__ params) {
    __shared__ float red_t[256];
    __shared__ float red_l[256];
    const int tid = threadIdx.x;
    float tacc = 0.0f, lacc = 0.0f;
    for (int i = tid; i < DLAT; i += 256) {
        float lc = log_c[i];
        float sp = (lc > 20.0f) ? lc : log1pf(expf(lc));
        float c  = sp + 1e-4f;
        float inv_c2 = 1.0f / (c * c);
        float bi = b[i];
        w1[i] = inv_c2;
        w2[i] = bi * inv_c2;
        tacc += bi * bi * inv_c2;
        lacc += 2.0f * logf(c);
    }
    red_t[tid] = tacc;
    red_l[tid] = lacc;
    __syncthreads();
    for (int off = 128; off > 0; off >>= 1) {
        if (tid < off) {
            red_t[tid] += red_t[tid + off];
            red_l[tid] += red_l[tid + off];
        }
        __syncthreads();
    }
    if (tid == 0) {
        float t   = red_t[0];
        float sld = red_l[0];
        const float LOG_2PI = 1.8378770664093453f;
        params[0] = 1.0f / (1.0f + t);
        params[1] = (float)DLAT * LOG_2PI + sld + log1pf(t);
    }
}

// ---------------------------------------------------------------------------
// Kernel 2: partial Mahalanobis sums. Block = 16 samples x 512 k (8 waves).
// Double-buffered async global->LDS staging; WMMA row-sum reduction.
// ---------------------------------------------------------------------------
__global__ __launch_bounds__(256) void nagvac_partial(const float* __restrict__ theta,
                                                      const float* __restrict__ mu,
                                                      const float* __restrict__ w1,
                                                      const float* __restrict__ w2,
                                                      float* __restrict__ partials) {
    __shared__ float tile[2][16 * LDS_STRIDE];   // 2 x 8448 B
    __shared__ float red[8][16][2];

    const int tid  = threadIdx.x;
    const int lane = tid & 31;                   // wave32
    const int wave = tid >> 5;
    const int m    = lane & 15;
    const int kh   = lane >> 4;
    const int row0   = blockIdx.x << 4;          // sample tile
    const int kbase0 = blockIdx.y * KPC;         // k chunk

    // B matrix: column 0 = ones -> row sum; replicated across both B VGPRs.
    const float bv = (m == 0) ? 1.0f : 0.0f;
    v2f bones; bones[0] = bv; bones[1] = bv;

    v8f accA = {};
    v8f accB = {};

    // Pipeline: issue t0,t1; {wait(2); compute t; issue t+2} ; drain.
    stage_tile(theta, tile[0], tid, row0, kbase0 + 0 * KC);
    stage_tile(theta, tile[1], tid, row0, kbase0 + 1 * KC);

    ASYNC_WAIT(2);                    // tile 0 landed (in-order completion)
    __syncthreads();
    compute_tile(tile[0], mu, w1, w2, m, kh, wave, kbase0 + 0 * KC, bones, accA, accB);
    __syncthreads();
    stage_tile(theta, tile[0], tid, row0, kbase0 + 2 * KC);

    ASYNC_WAIT(2);                    // tile 1 landed
    __syncthreads();
    compute_tile(tile[1], mu, w1, w2, m, kh, wave, kbase0 + 1 * KC, bones, accA, accB);
    __syncthreads();
    stage_tile(theta, tile[1], tid, row0, kbase0 + 3 * KC);

    ASYNC_WAIT(2);                    // tile 2 landed
    __syncthreads();
    compute_tile(tile[0], mu, w1, w2, m, kh, wave, kbase0 + 2 * KC, bones, accA, accB);
    __syncthreads();

    ASYNC_WAIT(0);                    // tile 3 landed
    __syncthreads();
    compute_tile(tile[1], mu, w1, w2, m, kh, wave, kbase0 + 3 * KC, bones, accA, accB);

    // Column N=0 lives in lane 0 (M=0..7) and lane 16 (M=8..15).
    if (m == 0) {
#pragma unroll
        for (int r = 0; r < 8; ++r) {
            red[wave][kh * 8 + r][0] = accA[r];
            red[wave][kh * 8 + r][1] = accB[r];
        }
    }
    __syncthreads();

    if (tid < 16) {
        float A = 0.0f, B = 0.0f;
#pragma unroll
        for (int w = 0; w < 8; ++w) {
            A += red[w][tid][0];
            B += red[w][tid][1];
        }
        float* p = partials + ((size_t)(blockIdx.x * KCHUNKS + blockIdx.y) * 16 + tid) * 2;
        p[0] = A;
        p[1] = B;
    }
}

// ---------------------------------------------------------------------------
// Kernel 3: deterministic finalize. out[s] = -0.5*(A - B^2/(1+t) + const).
// ---------------------------------------------------------------------------
__global__ __launch_bounds__(256) void nagvac_final(const float* __restrict__ partials,
                                                    const float* __restrict__ params,
                                                    float* __restrict__ out) {
    const int s = blockIdx.x * 256 + threadIdx.x;   // 0..1023
    const int tile16 = s >> 4;
    const int m      = s & 15;
    float A = 0.0f, B = 0.0f;
#pragma unroll
    for (int c = 0; c < KCHUNKS; ++c) {
        const float* p = partials + ((size_t)(tile16 * KCHUNKS + c) * 16 + m) * 2;
        A += p[0];
        B += p[1];
    }
    out[s] = -0.5f * (A - B * B * params[0] + params[1]);
}

// ---------------------------------------------------------------------------
extern "C" void kernel_launch(void* const* d_in, const int* in_sizes, int n_in,
                              void* d_out, int out_size, void* d_ws, size_t ws_size,
                              hipStream_t stream) {
    const float* theta = (const float*)d_in[0];   // [S, D]
    const float* mu    = (const float*)d_in[1];   // [D]
    const float* b     = (const float*)d_in[2];   // [D]
    const float* log_c = (const float*)d_in[3];   // [D]
    float* out = (float*)d_out;                   // [S]

    float* ws       = (float*)d_ws;
    float* w1       = ws;                          // [D]
    float* w2       = ws + DLAT;                   // [D]
    float* params   = ws + 2 * DLAT;               // [2]
    float* partials = ws + 2 * DLAT + 2;           // [64*8*16*2] = 16384 floats

    nagvac_prep<<<1, 256, 0, stream>>>(b, log_c, w1, w2, params);
    dim3 grid(S_SAMPLES / 16, KCHUNKS);
    nagvac_partial<<<grid, 256, 0, stream>>>(theta, mu, w1, w2, partials);
    nagvac_final<<<S_SAMPLES / 256, 256, 0, stream>>>(partials, params, out);
}